// PureLMKAN_2D_Layer_66640712565398
// MI455X (gfx1250) — compile-verified
//
// PureLMKAN 2D layer forward for MI455X (gfx1250), compile-only tuning.
//
// Strategy (see round-0 analysis): data-dependent bilinear gather => VALU
// bound, not WMMA-expressible without a losing sort/densification. We:
//  * keep params as 32 slices of 72.25KB, DMA'd into LDS with the CDNA5
//    Tensor Data Mover (tensor_load_to_lds + s_wait_tensorcnt), double
//    buffered and overlapped with compute,
//  * gather 4 corners per (b,p) with ds_load_b64 (bank-conflict-free: lanes
//    stride 8B over a 256B row),
//  * do the math with v_pk_fma_f32 / v_pk_mul_f32 using VOP3P op_sel to
//    broadcast scalar weights (2 FLOP/lane/instr, no weight duplication),
//  * write output through padded LDS staging for coalesced 128B stores.
//
// Round-2 deltas: hot kernel first in file (asm visibility); cell offsets
// fetched 4-at-a-time via ds_load_b128; weight quad fetched as one b128.

#include <hip/hip_runtime.h>
#include <stdint.h>

typedef float  v2f  __attribute__((ext_vector_type(2)));
typedef float  v4f  __attribute__((ext_vector_type(4)));
typedef unsigned int u32x4 __attribute__((ext_vector_type(4)));
typedef int    i32x8 __attribute__((ext_vector_type(8)));
typedef int    i32x4 __attribute__((ext_vector_type(4)));

#define GRID_G      17
#define N_CELLS     16
#define N_PAIRS     32
#define ODIM        64
#define BATCH       262144
#define SLICE_ELEMS (GRID_G * GRID_G * ODIM)   /* 18496 f32 */
#define SLICE_BYTES (SLICE_ELEMS * 4)          /* 73984 B   */

// LDS layout (dynamic shared), all 16B aligned:
#define OFF_BUF0  0
#define OFF_BUF1  SLICE_BYTES                 /* 73984  */
#define OFF_SCRW  (2 * SLICE_BYTES)           /* 147968: 8 waves * 64 * 16B */
#define OFF_SCRO  (OFF_SCRW + 8 * 64 * 16)    /* 156160: 8 waves * 64 * 4B  */
#define OFF_BORD  (OFF_SCRO + 8 * 64 * 4)     /* 158208: 17 * 4B            */
#define OFF_INVW  (OFF_BORD + 80)             /* 158288: 16 * 4B            */
#define SMEM_BYTES (OFF_INVW + 80)            /* 158368 < 160KB => 2 blocks/WGP */

// ---------------------------------------------------------------------------
// TDM: DMA one contiguous 18496-element slice (global -> LDS).
// D# packing per CDNA5 ISA ch.8 (CDNA5 ISA). 1-D tile:
// data_size=4B, tensor_dim0=tile_dim0=18496, dim1=tile_dim1=1, no padding,
// no multicast, no iteration. Groups 2/3 (and the trailing 8-dword group of
// the 6-arg builtin) are zero: tensor is <= 2-D.
// ---------------------------------------------------------------------------
__device__ __forceinline__ void tdm_load_slice(uint32_t lds_byte_addr,
                                               const float* gsrc) {
  uint64_t ga = (uint64_t)(uintptr_t)gsrc;
  u32x4 g0;
  g0[0] = 1u;                                   // count=1, is_restore=0, no gather
  g0[1] = lds_byte_addr;                        // lds_addr[31:0]
  g0[2] = (uint32_t)ga;                         // global_addr[31:0]
  g0[3] = (uint32_t)((ga >> 32) & 0x1FFFFFFu)   // global_addr[56:32]
        | (2u << 30);                           // type=2 ("image")
  i32x8 g1;
  g1[0] = (int)(2u << 16);                      // data_size = 2 (4 bytes)
  g1[1] = (int)((uint32_t)SLICE_ELEMS << 16);   // tensor_dim0[15:0] @ bits[63:48]
  g1[2] = (int)(((uint32_t)SLICE_ELEMS >> 16)   // tensor_dim0[31:16]
        | (1u << 16));                          // tensor_dim1 = 1
  g1[3] = (int)((uint32_t)SLICE_ELEMS << 16);   // tile_dim0 = 18496 (fits u16)
  g1[4] = (int)1u;                              // tile_dim1 = 1, tile_dim2 = 0
  g1[5] = (int)(uint32_t)SLICE_ELEMS;           // tensor_dim0_stride lo
  g1[6] = (int)(((uint32_t)SLICE_ELEMS & 0xFFFFu) << 16); // dim1_stride lo (unused)
  g1[7] = 0;
  i32x4 z4 = {0, 0, 0, 0};
  i32x8 z8 = {0, 0, 0, 0, 0, 0, 0, 0};
  // 6-arg toolchain variant: (g0, g1, g2, g3, g4, cpol)
  __builtin_amdgcn_tensor_load_to_lds(g0, g1, z4, z4, z8, 0);
}

// ---------------------------------------------------------------------------
// Main forward kernel (placed first so the disasm snippet shows it).
// Block = 256 threads (8 waves). Each wave owns 64 batch elements; lanes map
// to the output dim (lane l holds o = 2l, 2l+1 as a float2 accumulator).
// p-loop outer with double-buffered TDM slice loads.
// ---------------------------------------------------------------------------
__global__ __launch_bounds__(256) void lmkan_forward(
    const float* __restrict__ x,        // (64, BATCH)
    const float* __restrict__ borders,  // (17)
    const float* __restrict__ ft,       // (32, 17, 17, 64) transposed params
    float* __restrict__ out) {          // (64, BATCH)
  extern __shared__ char smem[];
  const int tid  = threadIdx.x;
  const int lane = tid & 31;
  const int wave = tid >> 5;

  float*    bordLds = (float*)(smem + OFF_BORD);
  float*    invwLds = (float*)(smem + OFF_INVW);
  float*    scrW    = (float*)(smem + OFF_SCRW + wave * 64 * 16);
  uint32_t* scrO    = (uint32_t*)(smem + OFF_SCRO + wave * 64 * 4);

  if (tid < GRID_G)  bordLds[tid] = borders[tid];
  if (tid < N_CELLS) invwLds[tid] = 1.0f / (borders[tid + 1] - borders[tid]);

  float br[GRID_G];
#pragma unroll
  for (int k = 0; k < GRID_G; ++k) br[k] = borders[k];   // uniform -> SGPRs

  const uint32_t ldsBase = (uint32_t)(uintptr_t)smem;    // low 32 bits = LDS offset
  if (wave == 0) {
    tdm_load_slice(ldsBase + OFF_BUF0, ft);
    __builtin_amdgcn_s_wait_tensorcnt(0);
  }
  __syncthreads();

  const int bBase = blockIdx.x * 512 + wave * 64;

  v2f acc[64];
#pragma unroll
  for (int k = 0; k < 64; ++k) acc[k] = (v2f)0.0f;

  for (int p = 0; p < N_PAIRS; ++p) {
    // Prefetch next slice into the other buffer (overlapped with compute).
    if (wave == 0 && p + 1 < N_PAIRS) {
      tdm_load_slice(ldsBase + (((p + 1) & 1) ? OFF_BUF1 : OFF_BUF0),
                     ft + (size_t)(p + 1) * SLICE_ELEMS);
    }

    // ---- Phase 1: lane-parallel binning/weights for 64 batch elems --------
#pragma unroll
    for (int pass = 0; pass < 2; ++pass) {
      const int b = bBase + pass * 32 + lane;
      float x1 = x[(size_t)(2 * p) * BATCH + b];
      float x2 = x[(size_t)(2 * p + 1) * BATCH + b];
      x1 = fminf(fmaxf(x1, br[0]), br[16]);
      x2 = fminf(fmaxf(x2, br[0]), br[16]);
      int i1 = 0, i2 = 0;   // branchless searchsorted(right)-1, clamped to 15
#pragma unroll
      for (int k = 1; k <= 15; ++k) {
        i1 += (x1 >= br[k]);
        i2 += (x2 >= br[k]);
      }
      const float a  = (x1 - bordLds[i1]) * invwLds[i1];
      const float bb = (x2 - bordLds[i2]) * invwLds[i2];
      const int slot = pass * 32 + lane;
      scrO[slot] = (uint32_t)(i1 * 4352 + i2 * 256);     // cell byte offset
      v4f wv = {1.0f - bb, bb, 1.0f - a, a};             // {B2, A2}
      ((v4f*)scrW)[slot] = wv;
    }
    // scratch is per-wave; same-wave LDS RAW handled by compiler dscnt waits.

    // ---- Phase 2: gather + packed bilinear FMA ----------------------------
    const char* buf = smem + ((p & 1) ? OFF_BUF1 : OFF_BUF0);
    const u32x4* scrO4 = (const u32x4*)scrO;
    const v4f*   scrW4 = (const v4f*)scrW;
#pragma unroll
    for (int k4 = 0; k4 < 16; ++k4) {
      const u32x4 off4 = scrO4[k4];        // 4 cell offsets: one ds_load_b128
#pragma unroll
      for (int u = 0; u < 4; ++u) {
        const int k = k4 * 4 + u;
        v4f W = scrW4[k];                  // {1-b, b, 1-a, a}: one ds_load_b128
        v2f B2 = W.xy;                     // {1-b, b}
        v2f A2 = W.zw;                     // {1-a, a}
        const v2f* cell = (const v2f*)(buf + off4[u]);
        v2f f00 = cell[lane];              // (i  , j  ), o = 2l..2l+1
        v2f f01 = cell[lane + 32];         // (i  , j+1)  +256B
        v2f f10 = cell[lane + 544];        // (i+1, j  )  +4352B
        v2f f11 = cell[lane + 576];        // (i+1, j+1)  +4608B
        v2f w0, w1;
        // w0 = (1-a) * {1-b, b} = {w00, w01}; broadcast A2.lo via op_sel
        asm("v_pk_mul_f32 %0, %1, %2 op_sel:[0,0] op_sel_hi:[0,1]"
            : "=v"(w0) : "v"(A2), "v"(B2));
        // w1 = a * {1-b, b} = {w10, w11}; broadcast A2.hi
        asm("v_pk_mul_f32 %0, %1, %2 op_sel:[1,0] op_sel_hi:[1,1]"
            : "=v"(w1) : "v"(A2), "v"(B2));
        // acc += f * broadcast(w.lo / w.hi)
        asm("v_pk_fma_f32 %0, %1, %2, %0 op_sel:[0,0,0] op_sel_hi:[1,0,1]"
            : "+v"(acc[k]) : "v"(f00), "v"(w0));
        asm("v_pk_fma_f32 %0, %1, %2, %0 op_sel:[0,1,0] op_sel_hi:[1,1,1]"
            : "+v"(acc[k]) : "v"(f01), "v"(w0));
        asm("v_pk_fma_f32 %0, %1, %2, %0 op_sel:[0,0,0] op_sel_hi:[1,0,1]"
            : "+v"(acc[k]) : "v"(f10), "v"(w1));
        asm("v_pk_fma_f32 %0, %1, %2, %0 op_sel:[0,1,0] op_sel_hi:[1,1,1]"
            : "+v"(acc[k]) : "v"(f11), "v"(w1));
      }
    }

    if (wave == 0) __builtin_amdgcn_s_wait_tensorcnt(0);  // next slice landed
    __syncthreads();  // everyone done reading buf[p&1]; safe to reuse next iter
  }

  // ---- Writeback: transpose through padded LDS staging for coalescing -----
  // Per-wave private 8448B region inside BUF0 (free after the final barrier).
  float* stage = (float*)(smem + wave * 8448);   // [32 b][66 f32] padded rows
#pragma unroll
  for (int c = 0; c < 2; ++c) {
#pragma unroll
    for (int k = 0; k < 32; ++k) {
      ((v2f*)(stage + k * 66))[lane] = acc[c * 32 + k];   // banks all distinct
    }
#pragma unroll
    for (int oo = 0; oo < ODIM; oo += 2) {
      v2f v = *(const v2f*)(stage + lane * 66 + oo);      // b in lanes now
      const int gb = bBase + c * 32 + lane;
      out[(size_t)(oo)     * BATCH + gb] = v.x;           // 128B coalesced
      out[(size_t)(oo + 1) * BATCH + gb] = v.y;
    }
  }
}

// ---------------------------------------------------------------------------
// Kernel 2: transpose func_parameter (i,j,o,p) -> Ft (p,i,j,o) so each p-slice
// is a contiguous 72.25KB block with o innermost (coalesced LDS gathers).
// ---------------------------------------------------------------------------
__global__ __launch_bounds__(256) void transpose_params(
    const float* __restrict__ src, float* __restrict__ dst) {
  int tid = blockIdx.x * 256 + threadIdx.x;            // 2312*256 == 591872 exact
  int o  = tid & 63;
  int t  = tid >> 6;
  int j  = t % GRID_G;
  int t2 = t / GRID_G;
  int i  = t2 % GRID_G;
  int p  = t2 / GRID_G;
  dst[tid] = src[((size_t)(i * GRID_G + j) * ODIM + o) * N_PAIRS + p];
}

// ---------------------------------------------------------------------------
extern "C" void kernel_launch(void* const* d_in, const int* in_sizes, int n_in,
                              void* d_out, int out_size, void* d_ws, size_t ws_size,
                              hipStream_t stream) {
  const float* fparam  = (const float*)d_in[0];   // (17,17,64,32)
  const float* x       = (const float*)d_in[1];   // (64, 262144)
  const float* borders = (const float*)d_in[2];   // (17)
  float*       out     = (float*)d_out;           // (64, 262144)
  float*       ft      = (float*)d_ws;            // 2.37MB transposed params

  transpose_params<<<2312, 256, 0, stream>>>(fparam, ft);

  (void)hipFuncSetAttribute(reinterpret_cast<const void*>(lmkan_forward),
                            hipFuncAttributeMaxDynamicSharedMemorySize,
                            SMEM_BYTES);
  lmkan_forward<<<512, 256, SMEM_BYTES, stream>>>(x, borders, ft, out);
}